// AEloss_11708080848916
// MI455X (gfx1250) — compile-verified
//
#include <hip/hip_runtime.h>
#include <math.h>

typedef __attribute__((ext_vector_type(2))) float v2f;
typedef __attribute__((ext_vector_type(8))) float v8f;

// AE (associative-embedding) loss: one wave32 per image (B=32 blocks).
// Phase 1: lane p (<30) owns person p; 17 independent b128 gathers (max MLP),
//          masked sums -> mean, n2, pull.
// Phase 2: diff[i,j] = n2_i + n2_j - 2 m_i.m_j computed fully inside chained
//          V_WMMA_F32_16X16X4_F32 ops (rank-4 Gram part + rank-2 n2 part),
//          3 upper-triangular 16x16 tiles = 6 WMMAs. Branchless exp/mask tail.
__global__ __launch_bounds__(32) void ae_loss_kernel(
    const float* __restrict__ tags,   // [B, N, 4]
    const int*   __restrict__ kp,     // [B, 30, 17, 2] (idx, vis)
    float*       __restrict__ out,    // [B, 2] (push_term, pull_term)
    long long N)
{
  constexpr int P = 30, K = 17;
  constexpr float SQRT2 = 1.41421356237309515f;
  const int b    = blockIdx.x;
  const int lane = threadIdx.x;

  const float* tagsb = tags + (long long)b * N * 4;
  const int*   kpb   = kp   + (long long)b * P * K * 2;

  float s1x = 0.f, s1y = 0.f, s1z = 0.f, s1w = 0.f, s2 = 0.f, cnt = 0.f;

  if (lane < P) {
    const int* kpp = kpb + lane * (K * 2);
    int idx[K], visv[K];
#pragma unroll
    for (int k = 0; k < K; ++k) {        // contiguous keypoint loads
      idx[k]  = kpp[2 * k];
      visv[k] = kpp[2 * k + 1];
    }
    float4 tv[K];
#pragma unroll
    for (int k = 0; k < K; ++k)          // 17 independent random b128 gathers
      tv[k] = *reinterpret_cast<const float4*>(tagsb + (long long)idx[k] * 4);
#pragma unroll
    for (int k = 0; k < K; ++k) {
      if (visv[k] > 0) {
        s1x += tv[k].x; s1y += tv[k].y; s1z += tv[k].z; s1w += tv[k].w;
        s2  += tv[k].x * tv[k].x + tv[k].y * tv[k].y +
               tv[k].z * tv[k].z + tv[k].w * tv[k].w;
        cnt += 1.0f;
      }
    }
  }

  const bool  valid = cnt > 0.0f;
  const float safe  = fmaxf(cnt, 1.0f);
  const float inv   = 1.0f / safe;
  const float mx = s1x * inv, my = s1y * inv, mz = s1z * inv, mw = s1w * inv;
  const float n2 = mx * mx + my * my + mz * mz + mw * mw;
  const float s1n2 = s1x * s1x + s1y * s1y + s1z * s1z + s1w * s1w;
  // sum_k m*||t-mean||^2 = s2 - ||s1||^2/cnt ; pull_p = that / cnt
  const float pull_c = valid ? (s2 - s1n2 * inv) * inv : 0.0f;

  // Wave-uniform validity bitmask (lanes 30/31 have cnt==0 -> bit clear).
  const unsigned vmask = (unsigned)__ballot(valid);

  __shared__ float smean[32][4];
  __shared__ float sn2[32];
  smean[lane][0] = mx; smean[lane][1] = my;
  smean[lane][2] = mz; smean[lane][3] = mw;   // lanes 30/31 naturally write 0
  sn2[lane] = n2;
  __syncthreads();

  // f32 16x16x4 fragment layout (A and B^T share the same register image):
  //   lane L<16  -> row/col L, K={0,1};  lane L+16 -> row/col L, K={2,3}.
  const int  row = lane & 15;
  const int  kof = (lane >> 4) << 1;      // 0 or 2
  const bool lo  = lane < 16;

  v2f aM0, aM1, bM0, bM1;                 // +/- sqrt2 * mean fragments
  aM0.x = -SQRT2 * smean[row][kof];      aM0.y = -SQRT2 * smean[row][kof + 1];
  aM1.x = -SQRT2 * smean[row + 16][kof]; aM1.y = -SQRT2 * smean[row + 16][kof + 1];
  bM0.x =  SQRT2 * smean[row][kof];      bM0.y =  SQRT2 * smean[row][kof + 1];
  bM1.x =  SQRT2 * smean[row + 16][kof]; bM1.y =  SQRT2 * smean[row + 16][kof + 1];

  // Rank-2 fragments: A row i = [n2_i, 1, 0, 0]; B col j = [1, n2_j, 0, 0]^T.
  const float n2r0 = sn2[row], n2r1 = sn2[row + 16];
  v2f aN0, aN1, bN0, bN1;
  aN0.x = lo ? n2r0 : 0.f;  aN0.y = lo ? 1.f  : 0.f;
  aN1.x = lo ? n2r1 : 0.f;  aN1.y = lo ? 1.f  : 0.f;
  bN0.x = lo ? 1.f  : 0.f;  bN0.y = lo ? n2r0 : 0.f;
  bN1.x = lo ? 1.f  : 0.f;  bN1.y = lo ? n2r1 : 0.f;

  v8f zero = {0.f, 0.f, 0.f, 0.f, 0.f, 0.f, 0.f, 0.f};
  // diff tiles: D = (n2_i + n2_j) + (-2 m_i.m_j), chained through C.
  v8f d00 = __builtin_amdgcn_wmma_f32_16x16x4_f32(false, aM0, false, bM0, (short)0, zero, false, false);
  d00     = __builtin_amdgcn_wmma_f32_16x16x4_f32(false, aN0, false, bN0, (short)0, d00,  false, false);
  v8f d01 = __builtin_amdgcn_wmma_f32_16x16x4_f32(false, aM0, false, bM1, (short)0, zero, false, false);
  d01     = __builtin_amdgcn_wmma_f32_16x16x4_f32(false, aN0, false, bN1, (short)0, d01,  false, false);
  v8f d11 = __builtin_amdgcn_wmma_f32_16x16x4_f32(false, aM1, false, bM1, (short)0, zero, false, false);
  d11     = __builtin_amdgcn_wmma_f32_16x16x4_f32(false, aN1, false, bN1, (short)0, d11,  false, false);

  // 16x16 f32 D layout: vgpr v, lane<16 -> (M=v, N=lane); lane>=16 -> (M=v+8, N=lane-16)
  const int hi  = (lane >> 4) * 8;
  const int col = lane & 15;
  float push = 0.0f;

#pragma unroll
  for (int v = 0; v < 8; ++v) {           // tile (0,0): rows 0-15, cols 0-15
    const int gi = hi + v, gj = col;
    const float diff = d00[v];
    const bool ok = (gi < gj) & (((vmask >> gi) & 1u) != 0u) &
                    (((vmask >> gj) & 1u) != 0u) & (diff != 0.0f);
    const float e = __expf(-diff);
    push += ok ? e : 0.0f;
  }
#pragma unroll
  for (int v = 0; v < 8; ++v) {           // tile (0,1): rows 0-15, cols 16-31 (gi<gj always)
    const int gi = hi + v, gj = 16 + col;
    const float diff = d01[v];
    const bool ok = (((vmask >> gi) & 1u) != 0u) &
                    (((vmask >> gj) & 1u) != 0u) & (diff != 0.0f);
    const float e = __expf(-diff);
    push += ok ? e : 0.0f;
  }
#pragma unroll
  for (int v = 0; v < 8; ++v) {           // tile (1,1): rows 16-31, cols 16-31
    const int gi = 16 + hi + v, gj = 16 + col;
    const float diff = d11[v];
    const bool ok = (gi < gj) & (((vmask >> gi) & 1u) != 0u) &
                    (((vmask >> gj) & 1u) != 0u) & (diff != 0.0f);
    const float e = __expf(-diff);
    push += ok ? e : 0.0f;
  }

  float pull = pull_c;
  float nv   = valid ? 1.0f : 0.0f;
#pragma unroll
  for (int off = 16; off > 0; off >>= 1) {
    push += __shfl_xor(push, off, 32);
    pull += __shfl_xor(pull, off, 32);
    nv   += __shfl_xor(nv,   off, 32);
  }

  if (lane == 0) {
    out[b * 2 + 0] = push / ((nv - 1.0f) * nv + 1e-6f);
    out[b * 2 + 1] = pull / (nv + 1e-6f);
  }
}

extern "C" void kernel_launch(void* const* d_in, const int* in_sizes, int n_in,
                              void* d_out, int out_size, void* d_ws, size_t ws_size,
                              hipStream_t stream) {
  const float* tags = (const float*)d_in[0];
  const int*   kp   = (const int*)d_in[1];
  float*       out  = (float*)d_out;

  const int B = out_size / 2;                                      // 32
  const long long N = (long long)in_sizes[0] / ((long long)B * 4); // 278528

  ae_loss_kernel<<<B, 32, 0, stream>>>(tags, kp, out, N);
}